// DWReg2DDecode3D_10634339025476
// MI455X (gfx1250) — compile-verified
//
#include <hip/hip_runtime.h>

typedef __attribute__((ext_vector_type(2))) float v2f;
typedef __attribute__((ext_vector_type(8))) float v8f;

// ---------------------------------------------------------------------------
// 1. Bilinear grid sample: feat (16,256,4,4), uv (16,64,2) -> xg (16,64,256)
//    stored (b, p, c) row-major so p is the K dim of the upsample GEMM.
// ---------------------------------------------------------------------------
__global__ void grid_sample_kernel(const float* __restrict__ uv,
                                   const float* __restrict__ feat,
                                   float* __restrict__ xg) {
    int bp = blockIdx.x;          // 0..1023
    int b  = bp >> 6;
    int c  = threadIdx.x;         // 0..255
    float u = uv[bp * 2 + 0];
    float v = uv[bp * 2 + 1];
    float gx = fminf(fmaxf((u - 0.5f) * 2.0f, -1.0f), 1.0f);
    float gy = fminf(fmaxf((v - 0.5f) * 2.0f, -1.0f), 1.0f);
    float x = (gx + 1.0f) * 0.5f * 3.0f;   // [0,3]
    float y = (gy + 1.0f) * 0.5f * 3.0f;
    float x0f = floorf(x), y0f = floorf(y);
    int x0 = (int)x0f, y0 = (int)y0f;
    int x1 = x0 + 1, y1 = y0 + 1;
    float fx = x - x0f, fy = y - y0f;
    float wa = (1.0f - fx) * (1.0f - fy);
    float wb = (1.0f - fx) * fy;
    float wc = fx * (1.0f - fy);
    float wd = fx * fy;
    const float* fb = feat + ((long)(b * 256 + c)) * 16;
    auto at = [&](int yy, int xx) -> float {
        if (xx < 0 || xx > 3 || yy < 0 || yy > 3) return 0.0f;  // zero-pad
        return fb[yy * 4 + xx];
    };
    float r = at(y0, x0) * wa + at(y1, x0) * wb + at(y0, x1) * wc + at(y1, x1) * wd;
    xg[(long)bp * 256 + c] = r;
}

// ---------------------------------------------------------------------------
// 2. Upsample GEMM via WMMA f32 16x16x4:
//    Y[b,v,c] = sum_p U[v,p] * X[b,p,c]   U:(784,64)  X:(16,64,256)
//    grid (784/16, 256/128, 16), block 256 (8 waves, one 16-col tile each)
// ---------------------------------------------------------------------------
__global__ void upsample_wmma_kernel(const float* __restrict__ U,
                                     const float* __restrict__ X,
                                     float* __restrict__ Y) {
    const int CIN = 64, CINP = 68, COUT = 256, M = 784;
    __shared__ float As[16 * CINP];
    int v0 = blockIdx.x * 16;
    int b  = blockIdx.z;
    // stage A tile (16 x 64) into LDS
    for (int e = threadIdx.x; e < 16 * CIN; e += blockDim.x) {
        int r = e >> 6, ci = e & 63;
        As[r * CINP + ci] = U[(v0 + r) * CIN + ci];
    }
    __syncthreads();
    int lane = threadIdx.x & 31, wave = threadIdx.x >> 5;
    int lo = lane & 15, hi = lane >> 4;
    int co = (blockIdx.y * 8 + wave) * 16 + lo;
    const float* Xb = X + (long)b * CIN * COUT;
    v8f acc = {};
#pragma unroll
    for (int kk = 0; kk < CIN / 4; ++kk) {
        int kb = kk * 4 + hi * 2;
        v2f a, bf;
        a.x  = As[lo * CINP + kb];
        a.y  = As[lo * CINP + kb + 1];
        bf.x = Xb[kb * COUT + co];
        bf.y = Xb[(kb + 1) * COUT + co];
        acc = __builtin_amdgcn_wmma_f32_16x16x4_f32(false, a, false, bf,
                                                    (short)0, acc, false, false);
    }
    float* Yb = Y + ((long)b * M + v0) * COUT + co;
    const float* af = (const float*)&acc;
#pragma unroll
    for (int r = 0; r < 8; ++r)
        Yb[(long)(r + hi * 8) * COUT] = af[r];
}

// ---------------------------------------------------------------------------
// 3. Mesh up-pool: P[b,n,c] = sum_{j<3} X[b, col[3n+j], c] * val[3n+j]
// ---------------------------------------------------------------------------
__global__ void pool_kernel(const float* __restrict__ X,
                            const int* __restrict__ col,
                            const float* __restrict__ val,
                            float* __restrict__ P,
                            int Nin, int Nout, int C, long total) {
    long i = (long)blockIdx.x * blockDim.x + threadIdx.x;
    if (i >= total) return;
    int  c  = (int)(i % C);
    long nc = i / C;
    int  n  = (int)(nc % Nout);
    int  b  = (int)(nc / Nout);
    const float* Xb = X + (long)b * Nin * C;
    int j = 3 * n;
    float r = Xb[(long)col[j] * C + c] * val[j]
            + Xb[(long)col[j + 1] * C + c] * val[j + 1]
            + Xb[(long)col[j + 2] * C + c] * val[j + 2];
    P[i] = r;
}

// ---------------------------------------------------------------------------
// 4. Fused spiral-gather + depthwise + pointwise GEMM (+ReLU) via WMMA.
//    A tile (16 rows x CIN) built in LDS: As[r,ci] = sum_s P[b,idx[n,s],ci]*dw[ci,s]
//    then Y[b,n,co] = relu( A @ W )   W:(CIN,Cout)
//    CIN templated (256 or 128) so the K loop unrolls with literal LDS offsets.
//    grid (N/16, Cout/(16*wavesPerWg), B), block 32*wavesPerWg
// ---------------------------------------------------------------------------
template <int CIN, bool RELU>
__global__ void dsconv_wmma_kernel(const float* __restrict__ P,
                                   const int* __restrict__ idx,
                                   const float* __restrict__ dw,
                                   const float* __restrict__ W,
                                   float* __restrict__ Y,
                                   int N, int Cout) {
    constexpr int CINP = CIN + 4;            // pad: LDS column reads conflict-free
    __shared__ float As[16 * CINP];
    int n0 = blockIdx.x * 16;
    int b  = blockIdx.z;
    const float* Pb = P + (long)b * N * CIN;
    for (int e = threadIdx.x; e < 16 * CIN; e += blockDim.x) {
        int r = e / CIN, ci = e - r * CIN;
        const int* ip = idx + (n0 + r) * 9;
        float acc = 0.0f;
#pragma unroll
        for (int s = 0; s < 9; ++s)
            acc += Pb[(long)ip[s] * CIN + ci] * dw[ci * 9 + s];
        As[r * CINP + ci] = acc;
    }
    __syncthreads();
    int lane = threadIdx.x & 31, wave = threadIdx.x >> 5;
    int lo = lane & 15, hi = lane >> 4;
    int co = (blockIdx.y * (blockDim.x >> 5) + wave) * 16 + lo;
    v8f acc = {};
#pragma unroll 8
    for (int kk = 0; kk < CIN / 4; ++kk) {
        int kb = kk * 4 + hi * 2;
        v2f a, bf;
        a.x  = As[lo * CINP + kb];
        a.y  = As[lo * CINP + kb + 1];
        bf.x = W[kb * Cout + co];
        bf.y = W[(kb + 1) * Cout + co];
        acc = __builtin_amdgcn_wmma_f32_16x16x4_f32(false, a, false, bf,
                                                    (short)0, acc, false, false);
    }
    float* Yb = Y + ((long)b * N + n0) * Cout + co;
    const float* af = (const float*)&acc;
#pragma unroll
    for (int r = 0; r < 8; ++r) {
        float v = af[r];
        if (RELU) v = fmaxf(v, 0.0f);
        Yb[(long)(r + hi * 8) * Cout] = v;
    }
}

// ---------------------------------------------------------------------------
// 5. Head: out[b,n,j] = sum_c ( sum_s X[b,idx[n,s],c]*dwh[c,s] ) * pwh[c,j]
//    X:(16,12544,64), out:(16,12544,3). One block (64 threads) per (b,n).
// ---------------------------------------------------------------------------
__global__ void head_kernel(const float* __restrict__ X,
                            const int* __restrict__ idx,
                            const float* __restrict__ dwh,
                            const float* __restrict__ pwh,
                            float* __restrict__ out, int N) {
    __shared__ float t[64];
    int bn = blockIdx.x;
    int b = bn / N, n = bn - b * N;
    int c = threadIdx.x;                    // 0..63
    const float* Xb = X + (long)b * N * 64;
    const int* ip = idx + n * 9;
    float acc = 0.0f;
#pragma unroll
    for (int s = 0; s < 9; ++s)
        acc += Xb[(long)ip[s] * 64 + c] * dwh[c * 9 + s];
    t[c] = acc;
    __syncthreads();
    if (c < 3) {
        float o = 0.0f;
        for (int k = 0; k < 64; ++k) o += t[k] * pwh[k * 3 + c];
        out[((long)b * N + n) * 3 + c] = o;
    }
}

// ---------------------------------------------------------------------------
extern "C" void kernel_launch(void* const* d_in, const int* in_sizes, int n_in,
                              void* d_out, int out_size, void* d_ws, size_t ws_size,
                              hipStream_t stream) {
    const float* uv   = (const float*)d_in[0];
    const float* feat = (const float*)d_in[1];
    const float* up   = (const float*)d_in[2];
    const float* dw0  = (const float*)d_in[3];
    const float* pw0  = (const float*)d_in[4];
    const float* dw1  = (const float*)d_in[5];
    const float* pw1  = (const float*)d_in[6];
    const float* dw2  = (const float*)d_in[7];
    const float* pw2  = (const float*)d_in[8];
    const float* dw3  = (const float*)d_in[9];
    const float* pw3  = (const float*)d_in[10];
    const float* dwh  = (const float*)d_in[11];
    const float* pwh  = (const float*)d_in[12];
    const int*   sp0  = (const int*)d_in[13];
    const int*   col0 = (const int*)d_in[14];
    const float* val0 = (const float*)d_in[15];
    const int*   sp1  = (const int*)d_in[16];
    const int*   col1 = (const int*)d_in[17];
    const float* val1 = (const float*)d_in[18];
    const int*   sp2  = (const int*)d_in[19];
    const int*   col2 = (const int*)d_in[20];
    const float* val2 = (const float*)d_in[21];
    const int*   sp3  = (const int*)d_in[22];
    const int*   col3 = (const int*)d_in[23];
    const float* val3 = (const float*)d_in[24];

    const int B = 16;
    float* ws   = (float*)d_ws;
    float* xg   = ws;                           // 262,144
    float* bufA = ws + 262144;                  // 12,845,056
    float* bufB = bufA + 12845056;              // 12,845,056
    float* pool = bufB + 12845056;              // 25,690,112

    // 1. grid sample -> xg (16,64,256)
    grid_sample_kernel<<<dim3(B * 64), 256, 0, stream>>>(uv, feat, xg);
    // 2. upsample -> x1 (16,784,256) in bufA
    upsample_wmma_kernel<<<dim3(784 / 16, 2, B), 256, 0, stream>>>(up, xg, bufA);

    // Stage 0: pool 784->1568 (C=256) from bufA; dsconv 256->256 -> bufB
    {
        long total = (long)B * 1568 * 256;
        pool_kernel<<<dim3((unsigned)((total + 255) / 256)), 256, 0, stream>>>(
            bufA, col3, val3, pool, 784, 1568, 256, total);
        dsconv_wmma_kernel<256, true><<<dim3(1568 / 16, 2, B), 256, 0, stream>>>(
            pool, sp3, dw0, pw0, bufB, 1568, 256);
    }
    // Stage 1: pool 1568->3136 (C=256) from bufB; dsconv 256->256 -> bufA
    {
        long total = (long)B * 3136 * 256;
        pool_kernel<<<dim3((unsigned)((total + 255) / 256)), 256, 0, stream>>>(
            bufB, col2, val2, pool, 1568, 3136, 256, total);
        dsconv_wmma_kernel<256, true><<<dim3(3136 / 16, 2, B), 256, 0, stream>>>(
            pool, sp2, dw1, pw1, bufA, 3136, 256);
    }
    // Stage 2: pool 3136->6272 (C=256) from bufA; dsconv 256->128 -> bufB
    {
        long total = (long)B * 6272 * 256;
        pool_kernel<<<dim3((unsigned)((total + 255) / 256)), 256, 0, stream>>>(
            bufA, col1, val1, pool, 3136, 6272, 256, total);
        dsconv_wmma_kernel<256, true><<<dim3(6272 / 16, 1, B), 256, 0, stream>>>(
            pool, sp1, dw2, pw2, bufB, 6272, 128);
    }
    // Stage 3: pool 6272->12544 (C=128) from bufB; dsconv 128->64 -> bufA
    {
        long total = (long)B * 12544 * 128;
        pool_kernel<<<dim3((unsigned)((total + 255) / 256)), 256, 0, stream>>>(
            bufB, col0, val0, pool, 6272, 12544, 128, total);
        dsconv_wmma_kernel<128, true><<<dim3(12544 / 16, 1, B), 128, 0, stream>>>(
            pool, sp0, dw3, pw3, bufA, 12544, 64);
    }
    // head: bufA (16,12544,64) -> d_out (16,12544,3)
    head_kernel<<<dim3(B * 12544), 64, 0, stream>>>(bufA, sp0, dwh, pwh,
                                                    (float*)d_out, 12544);
}